// FeatureInteraction_12532714570081
// MI455X (gfx1250) — compile-verified
//
#include <hip/hip_runtime.h>

typedef float v2f __attribute__((ext_vector_type(2)));
typedef float v8f __attribute__((ext_vector_type(8)));

#define NSPARSE 26
#define NFEAT   27      // 1 dense + 26 sparse
#define EMB     128
#define NOUT    351     // 27*26/2
#define WAVES_PER_BLK 4

__device__ __forceinline__ void store_triu(float* __restrict__ ob, int i, int j, float v) {
    // strict upper triangle of 27x27, row-major packed
    if (j > i && j < NFEAT && i < NFEAT) {
        int idx = 26 * i - (i * (i - 1)) / 2 + (j - i - 1);
        ob[idx] = v;
    }
}

__global__ __launch_bounds__(WAVES_PER_BLK * 32)
void feature_interaction_wmma(const float* __restrict__ dense,
                              const float* __restrict__ sparse,
                              float* __restrict__ out,
                              int nbatch) {
    // one wave = one batch; 27x128 fp32 tile staged in LDS per wave
    __shared__ float tile[WAVES_PER_BLK][NFEAT * EMB];

    const int wave = threadIdx.x >> 5;
    const int lane = threadIdx.x & 31;
    const int b    = blockIdx.x * WAVES_PER_BLK + wave;
    if (b >= nbatch) return;

    float* __restrict__ T = &tile[wave][0];

    // ---- stage batch b into LDS: 864 float4 total, 27 per lane, coalesced ----
    const float4* __restrict__ dsrc = (const float4*)(dense  + (size_t)b * EMB);
    const float4* __restrict__ ssrc = (const float4*)(sparse + (size_t)b * NSPARSE * EMB);
    #pragma unroll
    for (int it = 0; it < NFEAT; ++it) {
        int f = it * 32 + lane;                    // 0..863 flat float4 index
        float4 v = (f < (EMB / 4)) ? dsrc[f] : ssrc[f - (EMB / 4)];
        *(float4*)&T[f * 4] = v;
    }
    __syncthreads();

    // ---- Gram matrix via V_WMMA_F32_16X16X4_F32, K accumulated in steps of 4 ----
    // A (16x4) operand layout == B (4x16) operand layout for the same row block:
    //   lanes 0-15 : row (lane&15), K = k0, k0+1   (components .x,.y)
    //   lanes 16-31: row (lane&15), K = k0+2, k0+3
    const int half = lane >> 4;                    // 0 or 1
    const int r0   = lane & 15;                    // block0 row: 0..15
    int r1 = r0 + 16;                              // block1 row: 16..31
    if (r1 > NFEAT - 1) r1 = NFEAT - 1;            // clamp (results discarded by triu mask)

    v8f acc00 = {};   // rows 0-15  x cols 0-15
    v8f acc01 = {};   // rows 0-15  x cols 16-31
    v8f acc11 = {};   // rows 16-31 x cols 16-31

    #pragma unroll
    for (int kk = 0; kk < EMB / 4; ++kk) {
        const int k = kk * 4 + half * 2;
        v2f a0 = *(const v2f*)&T[r0 * EMB + k];    // ds_load_b64
        v2f a1 = *(const v2f*)&T[r1 * EMB + k];    // ds_load_b64
        acc00 = __builtin_amdgcn_wmma_f32_16x16x4_f32(false, a0, false, a0,
                                                      (short)0, acc00, false, false);
        acc01 = __builtin_amdgcn_wmma_f32_16x16x4_f32(false, a0, false, a1,
                                                      (short)0, acc01, false, false);
        acc11 = __builtin_amdgcn_wmma_f32_16x16x4_f32(false, a1, false, a1,
                                                      (short)0, acc11, false, false);
    }

    // ---- scatter strict-upper-triangle; C/D layout: lane<16 -> M=r, lane>=16 -> M=r+8 ----
    float* __restrict__ ob = out + (size_t)b * NOUT;
    const int nloc = lane & 15;
    #pragma unroll
    for (int r = 0; r < 8; ++r) {
        const int mloc = r + half * 8;
        store_triu(ob, mloc,      nloc,      acc00[r]);
        store_triu(ob, mloc,      nloc + 16, acc01[r]);
        store_triu(ob, mloc + 16, nloc + 16, acc11[r]);
    }
}

extern "C" void kernel_launch(void* const* d_in, const int* in_sizes, int n_in,
                              void* d_out, int out_size, void* d_ws, size_t ws_size,
                              hipStream_t stream) {
    const float* dense  = (const float*)d_in[0];   // (B, 1, 128) fp32
    const float* sparse = (const float*)d_in[1];   // (B, 26, 128) fp32
    float* out = (float*)d_out;                    // (B, 351) fp32

    const int nbatch = in_sizes[0] / EMB;          // B from dense's flat size
    const int nblocks = (nbatch + WAVES_PER_BLK - 1) / WAVES_PER_BLK;
    feature_interaction_wmma<<<nblocks, WAVES_PER_BLK * 32, 0, stream>>>(
        dense, sparse, out, nbatch);
}